// Mamba_30812095381960
// MI455X (gfx1250) — compile-verified
//
#include <hip/hip_runtime.h>
#include <hip/hip_bf16.h>
#include <math.h>

typedef __attribute__((ext_vector_type(2))) float v2f;
typedef __attribute__((ext_vector_type(8))) float v8f;

// ---------------------------------------------------------------------------
// Embedding gather: out[r, c] = emb[ids[r], c]
// ---------------------------------------------------------------------------
__global__ void embed_kernel(const int* __restrict__ ids,
                             const float* __restrict__ emb,
                             float* __restrict__ out, int rows, int dm) {
  int i = blockIdx.x * blockDim.x + threadIdx.x;
  if (i >= rows * dm) return;
  int r = i / dm, c = i - r * dm;
  out[i] = emb[(long)ids[r] * dm + c];
}

// ---------------------------------------------------------------------------
// RMSNorm over last dim (dm), one block per row.
// ---------------------------------------------------------------------------
__global__ void rmsnorm_kernel(const float* __restrict__ x,
                               const float* __restrict__ w,
                               float* __restrict__ out, int dm) {
  int row = blockIdx.x;
  const float* xr = x + (long)row * dm;
  float ss = 0.f;
  for (int c = threadIdx.x; c < dm; c += blockDim.x) { float v = xr[c]; ss += v * v; }
  __shared__ float red[32];
  for (int off = 16; off > 0; off >>= 1) ss += __shfl_down(ss, off, 32);
  int wid = threadIdx.x >> 5, lane = threadIdx.x & 31;
  if (lane == 0) red[wid] = ss;
  __syncthreads();
  if (wid == 0) {
    float v = (lane < (int)(blockDim.x >> 5)) ? red[lane] : 0.f;
    for (int off = 16; off > 0; off >>= 1) v += __shfl_down(v, off, 32);
    if (lane == 0) red[0] = v;
  }
  __syncthreads();
  float inv = rsqrtf(red[0] / (float)dm + 1e-5f);
  for (int c = threadIdx.x; c < dm; c += blockDim.x)
    out[(long)row * dm + c] = xr[c] * inv * w[c];
}

// ---------------------------------------------------------------------------
// Causal depthwise conv (K=4) + bias + SiLU.
// ---------------------------------------------------------------------------
__global__ void conv_silu_kernel(const float* __restrict__ xz,
                                 const float* __restrict__ w,
                                 const float* __restrict__ b,
                                 float* __restrict__ xc,
                                 int L, int dinner, int ldxz) {
  int d = blockIdx.x * blockDim.x + threadIdx.x;
  int batch = blockIdx.y;
  if (d >= dinner) return;
  float w0 = w[0 * dinner + d], w1 = w[1 * dinner + d];
  float w2 = w[2 * dinner + d], w3 = w[3 * dinner + d];
  float bias = b[d];
  float h0 = 0.f, h1 = 0.f, h2 = 0.f;  // x[l-3], x[l-2], x[l-1]
  const float* src = xz + (long)batch * L * ldxz + d;
  float* dst = xc + (long)batch * L * dinner + d;
  for (int l = 0; l < L; ++l) {
    float xv = src[(long)l * ldxz];
    float acc = w0 * h0 + w1 * h1 + w2 * h2 + w3 * xv + bias;
    dst[(long)l * dinner] = acc / (1.f + __expf(-acc));  // silu
    h0 = h1; h1 = h2; h2 = xv;
  }
}

// ---------------------------------------------------------------------------
// Selective SSM scan + skip + gating. One thread per (batch, channel).
// ---------------------------------------------------------------------------
__global__ void ssm_scan_kernel(const float* __restrict__ xc,
                                const float* __restrict__ xz,
                                const float* __restrict__ xdbl,
                                const float* __restrict__ delta,
                                const float* __restrict__ A_log,
                                const float* __restrict__ Dp,
                                float* __restrict__ y,
                                int L, int dinner, int ldxz, int lddbl) {
  int d = blockIdx.x * blockDim.x + threadIdx.x;
  int batch = blockIdx.y;
  float A[16];
#pragma unroll
  for (int n = 0; n < 16; ++n) A[n] = -__expf(A_log[(long)d * 16 + n]);
  float Dv = Dp[d];
  float s[16];
#pragma unroll
  for (int n = 0; n < 16; ++n) s[n] = 0.f;
  __shared__ float BC[32];
  const long rowbase = (long)batch * L;
  for (int l = 0; l < L; ++l) {
    long row = rowbase + l;
    if (threadIdx.x < 32) BC[threadIdx.x] = xdbl[row * lddbl + 48 + threadIdx.x];
    __syncthreads();
    float dl = delta[row * dinner + d];
    float xv = xc[row * dinner + d];
    float du = dl * xv;
    float acc = 0.f;
#pragma unroll
    for (int n = 0; n < 16; ++n) {
      float dA = __expf(dl * A[n]);
      s[n] = dA * s[n] + du * BC[n];
      acc += s[n] * BC[16 + n];
    }
    float res = xz[row * ldxz + dinner + d];
    float g = res / (1.f + __expf(-res));  // silu(res)
    y[row * dinner + d] = (acc + xv * Dv) * g;
    __syncthreads();
  }
}

// ---------------------------------------------------------------------------
// FP32 WMMA GEMM: C[M,N] = A[M,K] @ B[K,N]  (+ epilogue)
// Block: 128 threads = 4 waves. Block tile 128x64, K-tile 32.
// Each wave owns 32 rows x 64 cols: acc[2][4], 8 WMMAs per K-step-4 sharing
// 4 B-fragments (2x arithmetic intensity vs 16-row waves).
// Interior tiles staged via GLOBAL_LOAD_ASYNC_TO_LDS_B128 (ASYNCcnt);
// edge tiles / ragged K use guarded scalar staging.
// As stride 36 floats: 16B-aligned rows for b128 LDS writes and a
// conflict-free bank permutation for A-fragment reads (gcd(36,64)=4).
// ---------------------------------------------------------------------------
enum { EPI_NONE = 0, EPI_BIAS = 1, EPI_BIAS_SOFTPLUS = 2, EPI_RESIDUAL = 3 };

__device__ __forceinline__ unsigned lds_addr32(const void* p) {
  // Low 32 bits of a generic LDS pointer are the LDS byte address (ISA 10.2).
  return (unsigned)(uintptr_t)p;
}

template <int EPI>
__global__ void gemm_wmma_f32(const float* __restrict__ A,
                              const float* __restrict__ B,
                              const float* __restrict__ aux,  // bias[N] or residual[M,ldc]
                              float* __restrict__ C,
                              int M, int N, int K, int lda, int ldb, int ldc) {
  __shared__ float As[128][36];
  __shared__ float Bs[32][64];
  const int tid = threadIdx.x;
  const int wave = tid >> 5;
  const int lane = tid & 31;
  const int half = lane >> 4;    // selects K-pair within a fragment
  const int m16 = lane & 15;
  const int rowBase = blockIdx.y * 128;
  const int colBase = blockIdx.x * 64;

  const bool fast = (rowBase + 128 <= M) && (colBase + 64 <= N) &&
                    ((K & 31) == 0) && ((lda & 3) == 0) && ((ldb & 3) == 0);

  v8f acc[2][4] = {};

  for (int k0 = 0; k0 < K; k0 += 32) {
    if (fast) {
      // ---- async DMA staging: A = 8x b128, B = 4x b128 per thread ----
#pragma unroll
      for (int i = 0; i < 8; ++i) {
        int j = tid + i * 128;            // A tile: 128 rows x 8 float4
        int ra = j >> 3, ca = (j & 7) << 2;
        unsigned lA = lds_addr32(&As[ra][ca]);
        const float* gA = A + (size_t)(rowBase + ra) * lda + k0 + ca;
        asm volatile("global_load_async_to_lds_b128 %0, %1, off"
                     :: "v"(lA), "v"(gA) : "memory");
      }
#pragma unroll
      for (int i = 0; i < 4; ++i) {
        int j = tid + i * 128;            // B tile: 32 rows x 16 float4
        int rb = j >> 4, cb = (j & 15) << 2;
        unsigned lB = lds_addr32(&Bs[rb][cb]);
        const float* gB = B + (size_t)(k0 + rb) * ldb + colBase + cb;
        asm volatile("global_load_async_to_lds_b128 %0, %1, off"
                     :: "v"(lB), "v"(gB) : "memory");
      }
      if (k0 + 32 < K) {
        // warm L2 with the next K-tile while this one computes
        __builtin_prefetch(A + (size_t)(rowBase + tid) * lda + k0 + 32, 0, 1);
        __builtin_prefetch(B + (size_t)(k0 + 32 + (tid >> 2)) * ldb + colBase + ((tid & 3) << 4), 0, 1);
      }
      asm volatile("s_wait_asynccnt 0x0" ::: "memory");
    } else {
      // ---- guarded scalar staging (edge tiles / K not multiple of 32) ----
      for (int i = tid; i < 128 * 32; i += 128) {
        int r = i >> 5, c = i & 31;
        int gr = rowBase + r, gk = k0 + c;
        As[r][c] = (gr < M && gk < K) ? A[(long)gr * lda + gk] : 0.f;
      }
      for (int i = tid; i < 32 * 64; i += 128) {
        int r = i >> 6, c = i & 63;
        int gk = k0 + r, gc = colBase + c;
        Bs[r][c] = (gk < K && gc < N) ? B[(long)gk * ldb + gc] : 0.f;
      }
    }
    __syncthreads();

#pragma unroll
    for (int kk = 0; kk < 32; kk += 4) {
      int ks = kk + half * 2;
      v2f a0, a1;
      a0.x = As[wave * 32 + m16][ks];
      a0.y = As[wave * 32 + m16][ks + 1];
      a1.x = As[wave * 32 + 16 + m16][ks];
      a1.y = As[wave * 32 + 16 + m16][ks + 1];
#pragma unroll
      for (int t = 0; t < 4; ++t) {
        v2f b;
        b.x = Bs[ks][t * 16 + m16];
        b.y = Bs[ks + 1][t * 16 + m16];
        acc[0][t] = __builtin_amdgcn_wmma_f32_16x16x4_f32(
            false, a0, false, b, (short)0, acc[0][t], false, false);
        acc[1][t] = __builtin_amdgcn_wmma_f32_16x16x4_f32(
            false, a1, false, b, (short)0, acc[1][t], false, false);
      }
    }
    __syncthreads();
  }

  // Epilogue + store
#pragma unroll
  for (int r = 0; r < 2; ++r) {
#pragma unroll
    for (int t = 0; t < 4; ++t) {
#pragma unroll
      for (int i = 0; i < 8; ++i) {
        int m = rowBase + wave * 32 + r * 16 + i + half * 8;
        int n = colBase + t * 16 + m16;
        if (m < M && n < N) {
          float v = acc[r][t][i];
          if (EPI == EPI_BIAS) {
            v += aux[n];
          } else if (EPI == EPI_BIAS_SOFTPLUS) {
            v += aux[n];
            v = (v > 20.f) ? v : log1pf(__expf(v));  // softplus
          } else if (EPI == EPI_RESIDUAL) {
            v += aux[(long)m * ldc + n];
          }
          C[(long)m * ldc + n] = v;
        }
      }
    }
  }
}

// ---------------------------------------------------------------------------
// Host-side orchestration
// ---------------------------------------------------------------------------
extern "C" void kernel_launch(void* const* d_in, const int* in_sizes, int n_in,
                              void* d_out, int out_size, void* d_ws, size_t ws_size,
                              hipStream_t stream) {
  constexpr int Bz = 2, L = 1024, DM = 768, DI = 1536, DS = 16, DTR = 48;
  constexpr int NL = 2, V = 50264;
  constexpr int ROWS = Bz * L;         // 2048
  constexpr int LDXZ = 2 * DI;         // 3072
  constexpr int LDDBL = DTR + 2 * DS;  // 80

  const int*   ids        = (const int*)d_in[0];
  const float* emb        = (const float*)d_in[1];
  const float* in_proj_w  = (const float*)d_in[2];
  const float* conv_w     = (const float*)d_in[3];
  const float* conv_b     = (const float*)d_in[4];
  const float* x_proj_w   = (const float*)d_in[5];
  const float* dt_proj_w  = (const float*)d_in[6];
  const float* dt_proj_b  = (const float*)d_in[7];
  const float* A_log      = (const float*)d_in[8];
  const float* D_param    = (const float*)d_in[9];
  const float* out_proj_w = (const float*)d_in[10];
  const float* norm_w     = (const float*)d_in[11];
  const float* norm_f_w   = (const float*)d_in[12];
  const float* lm_head_w  = (const float*)d_in[13];
  const float* lm_head_b  = (const float*)d_in[14];

  float* ws = (float*)d_ws;
  size_t o = 0;
  float* x     = ws + o; o += (size_t)ROWS * DM;     // residual stream
  float* xn    = ws + o; o += (size_t)ROWS * DM;     // normed
  float* xz    = ws + o; o += (size_t)ROWS * LDXZ;   // in_proj output (xc_raw | res)
  float* xc    = ws + o; o += (size_t)ROWS * DI;     // conv+silu output
  float* xdbl  = ws + o; o += (size_t)ROWS * LDDBL;  // x_proj output (dt | B | C)
  float* delta = ws + o; o += (size_t)ROWS * DI;     // softplus(dt_proj)
  float* yb    = ws + o; o += (size_t)ROWS * DI;     // scan output

  dim3 gblk(128);
  const int MT = 128;  // GEMM M-tile

  embed_kernel<<<(ROWS * DM + 255) / 256, 256, 0, stream>>>(ids, emb, x, ROWS, DM);

  for (int i = 0; i < NL; ++i) {
    rmsnorm_kernel<<<ROWS, 256, 0, stream>>>(x, norm_w + (size_t)i * DM, xn, DM);

    // in_proj: (2048x768) @ (768x3072) -> xz
    gemm_wmma_f32<EPI_NONE><<<dim3((LDXZ + 63) / 64, (ROWS + MT - 1) / MT), gblk, 0, stream>>>(
        xn, in_proj_w + (size_t)i * DM * LDXZ, nullptr, xz,
        ROWS, LDXZ, DM, DM, LDXZ, LDXZ);

    // depthwise causal conv + silu
    conv_silu_kernel<<<dim3(DI / 256, Bz), 256, 0, stream>>>(
        xz, conv_w + (size_t)i * 4 * DI, conv_b + (size_t)i * DI, xc, L, DI, LDXZ);

    // x_proj: (2048x1536) @ (1536x80) -> xdbl
    gemm_wmma_f32<EPI_NONE><<<dim3((LDDBL + 63) / 64, (ROWS + MT - 1) / MT), gblk, 0, stream>>>(
        xc, x_proj_w + (size_t)i * DI * LDDBL, nullptr, xdbl,
        ROWS, LDDBL, DI, DI, LDDBL, LDDBL);

    // dt_proj + bias + softplus: (2048x48) @ (48x1536) -> delta
    gemm_wmma_f32<EPI_BIAS_SOFTPLUS><<<dim3((DI + 63) / 64, (ROWS + MT - 1) / MT), gblk, 0, stream>>>(
        xdbl, dt_proj_w + (size_t)i * DTR * DI, dt_proj_b + (size_t)i * DI, delta,
        ROWS, DI, DTR, LDDBL, DI, DI);

    // selective scan + skip + gating -> yb
    ssm_scan_kernel<<<dim3(DI / 256, Bz), 256, 0, stream>>>(
        xc, xz, xdbl, delta, A_log + (size_t)i * DI * DS, D_param + (size_t)i * DI,
        yb, L, DI, LDXZ, LDDBL);

    // out_proj + residual: (2048x1536) @ (1536x768) + x -> x
    gemm_wmma_f32<EPI_RESIDUAL><<<dim3((DM + 63) / 64, (ROWS + MT - 1) / MT), gblk, 0, stream>>>(
        yb, out_proj_w + (size_t)i * DI * DM, x, x,
        ROWS, DM, DI, DI, DM, DM);
  }

  rmsnorm_kernel<<<ROWS, 256, 0, stream>>>(x, norm_f_w, xn, DM);

  // lm_head: (2048x768) @ (768x50264) + bias -> d_out (fp32)
  gemm_wmma_f32<EPI_BIAS><<<dim3((V + 63) / 64, (ROWS + MT - 1) / MT), gblk, 0, stream>>>(
      xn, lm_head_w, lm_head_b, (float*)d_out,
      ROWS, V, DM, DM, V, V);
}